// SparseAttention_23605140259494
// MI455X (gfx1250) — compile-verified
//
#include <hip/hip_runtime.h>
#include <math.h>

// ---------------- problem constants ----------------
constexpr int B_    = 2;
constexpr int T_    = 4096;
constexpr int DIM   = 512;
constexpr int HEADS = 16;
constexpr int DH    = 64;
constexpr int BALL  = 256;
constexpr int CBS   = 32;
constexpr int STRIDE= 16;
constexpr int SBLK  = 64;
constexpr int NSEL  = 16;
constexpr int W_    = T_ / STRIDE;     // 256 compressed blocks
constexpr int NQ    = T_ / SBLK;       // 64 query blocks
constexpr int NB    = T_ / BALL;       // 16 balls
constexpr int CD    = CBS * DH;        // 2048 compress input dim
constexpr int INNER = HEADS * DH;      // 1024
constexpr int QKVN  = INNER + 2 * DH;  // 1152
constexpr int NKC   = W_ + 1;          // 257 compressed keys (1 mem slot)
constexpr float SCALE_ = 0.125f;       // DH^-0.5
constexpr float NEGBIG = -3.0e38f;

typedef __attribute__((ext_vector_type(2))) float v2f;
typedef __attribute__((ext_vector_type(8))) float v8f;

__device__ inline v8f vzero8() {
  v8f v;
  #pragma unroll
  for (int i = 0; i < 8; ++i) v[i] = 0.f;
  return v;
}

// D = A(16x4) * B(4x16) + C   -- f32 WMMA
__device__ inline v8f wmma4(v2f a, v2f b, v8f c) {
  return __builtin_amdgcn_wmma_f32_16x16x4_f32(false, a, false, b, (short)0, c,
                                               false, false);
}

// ---------------- generic fp32 WMMA GEMM ----------------
// C[M,N] = A[M,K] @ B[K,N]  (+bias[N]) (+relu). Row-major. 1 wave/block.
// grid = (N/64, M/16), block = 32. M%16==0, N%64==0, K%16==0.
template <int EPI>
__global__ void __launch_bounds__(32)
gemm_wmma(const float* __restrict__ A, const float* __restrict__ Bm,
          const float* __restrict__ bias, float* __restrict__ C,
          int M, int N, int K) {
  __shared__ float As[16][20];
  __shared__ float Bs[16][68];
  const int nt = blockIdx.x, mt = blockIdx.y;
  const int lane = threadIdx.x;
  const int mrow = lane & 15;
  const int khalf = (lane >> 4) * 2;
  v8f acc[4];
  #pragma unroll
  for (int n = 0; n < 4; ++n) acc[n] = vzero8();

  for (int k0 = 0; k0 < K; k0 += 16) {
    for (int i = lane; i < 256; i += 32) {
      int r = i >> 4, c = i & 15;
      As[r][c] = A[(size_t)(mt * 16 + r) * K + k0 + c];
    }
    for (int i = lane; i < 1024; i += 32) {
      int r = i >> 6, c = i & 63;
      Bs[r][c] = Bm[(size_t)(k0 + r) * N + nt * 64 + c];
    }
    __syncthreads();
    #pragma unroll
    for (int kk = 0; kk < 4; ++kk) {
      v2f a;
      a.x = As[mrow][kk * 4 + khalf];
      a.y = As[mrow][kk * 4 + khalf + 1];
      #pragma unroll
      for (int n = 0; n < 4; ++n) {
        v2f b;
        b.x = Bs[kk * 4 + khalf][n * 16 + mrow];
        b.y = Bs[kk * 4 + khalf + 1][n * 16 + mrow];
        acc[n] = wmma4(a, b, acc[n]);
      }
    }
    __syncthreads();
  }
  const int rh = (lane >> 4) * 8;
  #pragma unroll
  for (int n = 0; n < 4; ++n) {
    #pragma unroll
    for (int vr = 0; vr < 8; ++vr) {
      int r = mt * 16 + vr + rh;
      int c = nt * 64 + n * 16 + mrow;
      float v = acc[n][vr];
      if (EPI >= 1) v += bias[c];
      if (EPI == 2) v = fmaxf(v, 0.f);
      C[(size_t)r * N + c] = v;
    }
  }
}

// ---------------- RMSNorm ----------------
__global__ void rmsnorm_kernel(const float* __restrict__ inp,
                               const float* __restrict__ w,
                               float* __restrict__ x) {
  __shared__ float red[8];
  const size_t bt = blockIdx.x;
  const float* row = inp + bt * DIM;
  float ss = 0.f;
  for (int i = threadIdx.x; i < DIM; i += 256) { float v = row[i]; ss += v * v; }
  for (int off = 1; off < 32; off <<= 1) ss += __shfl_xor(ss, off, 32);
  if ((threadIdx.x & 31) == 0) red[threadIdx.x >> 5] = ss;
  __syncthreads();
  if (threadIdx.x == 0) {
    float t = 0.f;
    for (int i = 0; i < 8; ++i) t += red[i];
    red[0] = rsqrtf(t / (float)DIM + 1.1920929e-07f);
  }
  __syncthreads();
  const float r = red[0];
  for (int i = threadIdx.x; i < DIM; i += 256) x[bt * DIM + i] = row[i] * r * w[i];
}

// ---------------- gates = sigmoid(x @ comb_W + comb_b) ----------------
__global__ void gates_kernel(const float* __restrict__ x,
                             const float* __restrict__ combW,
                             const float* __restrict__ combB,
                             float* __restrict__ gates) {
  __shared__ float xs[DIM];
  const size_t bt = blockIdx.x;
  for (int i = threadIdx.x; i < DIM; i += 64) xs[i] = x[bt * DIM + i];
  __syncthreads();
  const int c = threadIdx.x;
  if (c < 3 * HEADS) {
    float s = combB[c];
    for (int d = 0; d < DIM; ++d) s += xs[d] * combW[d * (3 * HEADS) + c];
    gates[bt * (3 * HEADS) + c] = 1.f / (1.f + __expf(-s));
  }
}

// ---------------- build unfolded compress windows ----------------
// out[b*W + w][j*64+d] = kv[b, w*16 + j - 16, d]  (zero-pad t'<0)
__global__ void build_win(const float* __restrict__ qkv, float* __restrict__ out,
                          int kvoff) {
  const int bw = blockIdx.x;
  const int b = bw / W_, w = bw % W_;
  for (int i = threadIdx.x; i < CD; i += blockDim.x) {
    int j = i >> 6, d = i & 63;
    int t = w * STRIDE + j - (CBS - STRIDE);
    float v = 0.f;
    if (t >= 0) v = qkv[((size_t)b * T_ + t) * QKVN + kvoff + d];
    out[(size_t)bw * CD + i] = v;
  }
}

// ---------------- pack ck/cv with memory slot ----------------
__global__ void pack_ck(const float* __restrict__ ckt, const float* __restrict__ cvt,
                        const float* __restrict__ mem_kv,
                        float* __restrict__ ck, float* __restrict__ cv) {
  const int bn = blockIdx.x;
  const int b = bn / NKC, n = bn % NKC;
  const int d = threadIdx.x;
  size_t o = ((size_t)b * NKC + n) * DH + d;
  if (n == 0) { ck[o] = mem_kv[d]; cv[o] = mem_kv[DH + d]; }
  else {
    size_t s = ((size_t)b * W_ + (n - 1)) * DH + d;
    ck[o] = ckt[s]; cv[o] = cvt[s];
  }
}

// ---------------- shared flash-attention step (wave32) ----------------
__device__ inline void softmax_step(float sv[8], float m_old[8], float l[8],
                                    v8f O[4], float (*Ps)[20],
                                    const float (*Vs)[68], int lane) {
  const int mrow = lane & 15;
  const int hf = lane >> 4;
  const int khalf = hf * 2;
  float mt[8], p[8], rs[8];
  #pragma unroll
  for (int r = 0; r < 8; ++r) {
    float v = sv[r];
    for (int off = 1; off < 16; off <<= 1) v = fmaxf(v, __shfl_xor(v, off, 32));
    mt[r] = fmaxf(v, m_old[r]);
  }
  #pragma unroll
  for (int r = 0; r < 8; ++r) {
    rs[r] = __expf(m_old[r] - mt[r]);
    p[r] = __expf(sv[r] - mt[r]);
    float ps = p[r];
    for (int off = 1; off < 16; off <<= 1) ps += __shfl_xor(ps, off, 32);
    l[r] = l[r] * rs[r] + ps;
    m_old[r] = mt[r];
  }
  #pragma unroll
  for (int n = 0; n < 4; ++n) {
    #pragma unroll
    for (int r = 0; r < 8; ++r) O[n][r] *= rs[r];
  }
  __syncthreads();
  #pragma unroll
  for (int r = 0; r < 8; ++r) Ps[r + hf * 8][mrow] = p[r];
  __syncthreads();
  #pragma unroll
  for (int c = 0; c < 4; ++c) {
    v2f ap;
    ap.x = Ps[mrow][c * 4 + khalf];
    ap.y = Ps[mrow][c * 4 + khalf + 1];
    #pragma unroll
    for (int n = 0; n < 4; ++n) {
      v2f bv;
      bv.x = Vs[c * 4 + khalf][n * 16 + mrow];
      bv.y = Vs[c * 4 + khalf + 1][n * 16 + mrow];
      O[n] = wmma4(ap, bv, O[n]);
    }
  }
}

// ---------------- compressed-branch attention ----------------
// grid (T/16, HEADS, B), block 32. writes merged = gate0 * c_out
__global__ void __launch_bounds__(32)
attn_compressed(const float* __restrict__ qkv, const float* __restrict__ ck,
                const float* __restrict__ cv, const float* __restrict__ gates,
                float* __restrict__ merged) {
  __shared__ float Ks[16][68], Vs[16][68], Ps[16][20];
  const int qt = blockIdx.x, h = blockIdx.y, b = blockIdx.z;
  const int lane = threadIdx.x;
  const int mrow = lane & 15, hf = lane >> 4, khalf = hf * 2;
  v2f aq[16];
  {
    const float* qrow = qkv + ((size_t)b * T_ + qt * 16 + mrow) * QKVN + h * DH;
    #pragma unroll
    for (int c = 0; c < 16; ++c) { aq[c].x = qrow[c * 4 + khalf]; aq[c].y = qrow[c * 4 + khalf + 1]; }
  }
  v8f O[4]; float m_old[8], l[8];
  #pragma unroll
  for (int n = 0; n < 4; ++n) O[n] = vzero8();
  #pragma unroll
  for (int r = 0; r < 8; ++r) { m_old[r] = -1e30f; l[r] = 0.f; }

  const int NT = (NKC + 15) / 16;  // 17
  for (int kt = 0; kt < NT; ++kt) {
    for (int i = lane; i < 16 * 64; i += 32) {
      int r = i >> 6, d = i & 63;
      int key = kt * 16 + r;
      int kc = key < NKC ? key : NKC - 1;
      Ks[r][d] = ck[((size_t)b * NKC + kc) * DH + d];
      Vs[r][d] = cv[((size_t)b * NKC + kc) * DH + d];
    }
    __syncthreads();
    v8f s = vzero8();
    #pragma unroll
    for (int c = 0; c < 16; ++c) {
      v2f bf;
      bf.x = Ks[mrow][c * 4 + khalf];
      bf.y = Ks[mrow][c * 4 + khalf + 1];
      s = wmma4(aq[c], bf, s);
    }
    const bool colv = (kt * 16 + mrow) < NKC;
    float sv[8];
    #pragma unroll
    for (int r = 0; r < 8; ++r) sv[r] = colv ? s[r] * SCALE_ : NEGBIG;
    softmax_step(sv, m_old, l, O, Ps, Vs, lane);
    __syncthreads();
  }
  #pragma unroll
  for (int n = 0; n < 4; ++n) {
    #pragma unroll
    for (int vr = 0; vr < 8; ++vr) {
      int t = qt * 16 + vr + hf * 8;
      int d = n * 16 + mrow;
      float g = gates[((size_t)b * T_ + t) * (3 * HEADS) + h * 3 + 0];
      merged[((size_t)b * T_ + t) * INNER + h * DH + d] = g * (O[n][vr] / l[vr]);
    }
  }
}

// ---------------- importance + top-k selection ----------------
// grid = B*NQ (rows t = 0..63), block 64
__global__ void select_topk(const float* __restrict__ qkv,
                            const float* __restrict__ ck, int* __restrict__ sel) {
  __shared__ float qbar[DH];
  __shared__ float impf[NQ];
  const int bt = blockIdx.x;
  const int b = bt / NQ, t = bt % NQ;
  const int tid = threadIdx.x;
  if (tid < DH) {
    float s = 0.f;
    const float* row = qkv + ((size_t)b * T_ + t) * QKVN;
    for (int h = 0; h < HEADS; ++h) s += row[h * DH + tid];
    qbar[tid] = s * (1.0f / HEADS);
  }
  __syncthreads();
  {
    const int jf = tid;  // 64 fine blocks
    float s = 0.f;
    for (int r = 0; r < 4; ++r) {
      const float* ckr = ck + ((size_t)b * NKC + 1 + jf * 4 + r) * DH;
      for (int d = 0; d < DH; ++d) s += qbar[d] * ckr[d];
    }
    float v = s * (SCALE_ * 0.25f);
    if (jf == t / SBLK) v = NEGBIG;  // diagonal mask
    impf[jf] = v;
  }
  __syncthreads();
  if (tid == 0) {
    for (int k = 0; k < NSEL; ++k) {
      int best = 0; float bv = -3.39e38f;
      for (int j = 0; j < NQ; ++j)
        if (impf[j] > bv) { bv = impf[j]; best = j; }
      sel[((size_t)b * NQ + t) * NSEL + k] = best;
      impf[best] = -3.4e38f;
    }
  }
}

// ---------------- fine-branch attention (causal, selected blocks) ----------------
// grid (NQ*4, HEADS, B), block 32. merged += gate1 * fine
__global__ void __launch_bounds__(32)
attn_fine(const float* __restrict__ qkv, const int* __restrict__ sel,
          const float* __restrict__ gates, float* __restrict__ merged) {
  __shared__ float Ks[16][68], Vs[16][68], Ps[16][20];
  const int qb = blockIdx.x >> 2, qsub = blockIdx.x & 3;
  const int h = blockIdx.y, b = blockIdx.z;
  const int lane = threadIdx.x;
  const int mrow = lane & 15, hf = lane >> 4, khalf = hf * 2;
  const int qt0 = qb * SBLK + qsub * 16;
  v2f aq[16];
  {
    const float* qrow = qkv + ((size_t)b * T_ + qt0 + mrow) * QKVN + h * DH;
    #pragma unroll
    for (int c = 0; c < 16; ++c) { aq[c].x = qrow[c * 4 + khalf]; aq[c].y = qrow[c * 4 + khalf + 1]; }
  }
  v8f O[4]; float m_old[8], l[8];
  #pragma unroll
  for (int n = 0; n < 4; ++n) O[n] = vzero8();
  #pragma unroll
  for (int r = 0; r < 8; ++r) { m_old[r] = -1e30f; l[r] = 0.f; }

  for (int s0 = 0; s0 < NSEL; ++s0) {
    int sb = sel[((size_t)b * NQ + qb) * NSEL + s0];
    sb = sb < qb ? sb : qb;  // causal clamp
    for (int sub = 0; sub < 4; ++sub) {
      const int k0 = sb * SBLK + sub * 16;
      for (int i = lane; i < 16 * 64; i += 32) {
        int r = i >> 6, d = i & 63;
        const float* row = qkv + ((size_t)b * T_ + k0 + r) * QKVN;
        Ks[r][d] = row[INNER + d];
        Vs[r][d] = row[INNER + DH + d];
      }
      __syncthreads();
      v8f s = vzero8();
      #pragma unroll
      for (int c = 0; c < 16; ++c) {
        v2f bf;
        bf.x = Ks[mrow][c * 4 + khalf];
        bf.y = Ks[mrow][c * 4 + khalf + 1];
        s = wmma4(aq[c], bf, s);
      }
      const int kpos = k0 + mrow;
      float sv[8];
      #pragma unroll
      for (int r = 0; r < 8; ++r) {
        int qpos = qt0 + r + hf * 8;
        sv[r] = (kpos <= qpos) ? s[r] * SCALE_ : NEGBIG;
      }
      softmax_step(sv, m_old, l, O, Ps, Vs, lane);
      __syncthreads();
    }
  }
  #pragma unroll
  for (int n = 0; n < 4; ++n) {
    #pragma unroll
    for (int vr = 0; vr < 8; ++vr) {
      int t = qt0 + vr + hf * 8;
      int d = n * 16 + mrow;
      float g = gates[((size_t)b * T_ + t) * (3 * HEADS) + h * 3 + 1];
      merged[((size_t)b * T_ + t) * INNER + h * DH + d] += g * (O[n][vr] / l[vr]);
    }
  }
}

// ---------------- sliding / ball attention with distance bias ----------------
// grid (NB*16, HEADS, B), block 32. merged += gate2 * slide
__global__ void __launch_bounds__(32)
attn_slide(const float* __restrict__ qkv, const float* __restrict__ pos,
           const float* __restrict__ sigma, const float* __restrict__ gates,
           float* __restrict__ merged) {
  __shared__ float Ks[16][68], Vs[16][68], Ps[16][20];
  __shared__ float Qp[16][4], Kp[16][4];
  const int nb = blockIdx.x >> 4, qsub = blockIdx.x & 15;
  const int h = blockIdx.y, b = blockIdx.z;
  const int lane = threadIdx.x;
  const int mrow = lane & 15, hf = lane >> 4, khalf = hf * 2;
  const int qt0 = nb * BALL + qsub * 16;
  const float sg = sigma[h];
  v2f aq[16];
  {
    const float* qrow = qkv + ((size_t)b * T_ + qt0 + mrow) * QKVN + h * DH;
    #pragma unroll
    for (int c = 0; c < 16; ++c) { aq[c].x = qrow[c * 4 + khalf]; aq[c].y = qrow[c * 4 + khalf + 1]; }
  }
  for (int i = lane; i < 48; i += 32)
    Qp[i / 3][i % 3] = pos[((size_t)b * T_ + qt0 + i / 3) * 3 + (i % 3)];
  v8f O[4]; float m_old[8], l[8];
  #pragma unroll
  for (int n = 0; n < 4; ++n) O[n] = vzero8();
  #pragma unroll
  for (int r = 0; r < 8; ++r) { m_old[r] = -1e30f; l[r] = 0.f; }

  for (int kt = 0; kt < BALL / 16; ++kt) {
    const int k0 = nb * BALL + kt * 16;
    for (int i = lane; i < 16 * 64; i += 32) {
      int r = i >> 6, d = i & 63;
      const float* row = qkv + ((size_t)b * T_ + k0 + r) * QKVN;
      Ks[r][d] = row[INNER + d];
      Vs[r][d] = row[INNER + DH + d];
    }
    for (int i = lane; i < 48; i += 32)
      Kp[i / 3][i % 3] = pos[((size_t)b * T_ + k0 + i / 3) * 3 + (i % 3)];
    __syncthreads();
    v8f s = vzero8();
    #pragma unroll
    for (int c = 0; c < 16; ++c) {
      v2f bf;
      bf.x = Ks[mrow][c * 4 + khalf];
      bf.y = Ks[mrow][c * 4 + khalf + 1];
      s = wmma4(aq[c], bf, s);
    }
    const float kx = Kp[mrow][0], ky = Kp[mrow][1], kz = Kp[mrow][2];
    float sv[8];
    #pragma unroll
    for (int r = 0; r < 8; ++r) {
      int qr = r + hf * 8;
      float dx = Qp[qr][0] - kx, dy = Qp[qr][1] - ky, dz = Qp[qr][2] - kz;
      float dist = sqrtf(fmaxf(dx * dx + dy * dy + dz * dz, 0.f));
      sv[r] = s[r] * SCALE_ + sg * dist;
    }
    softmax_step(sv, m_old, l, O, Ps, Vs, lane);
    __syncthreads();
  }
  #pragma unroll
  for (int n = 0; n < 4; ++n) {
    #pragma unroll
    for (int vr = 0; vr < 8; ++vr) {
      int t = qt0 + vr + hf * 8;
      int d = n * 16 + mrow;
      float g = gates[((size_t)b * T_ + t) * (3 * HEADS) + h * 3 + 2];
      merged[((size_t)b * T_ + t) * INNER + h * DH + d] += g * (O[n][vr] / l[vr]);
    }
  }
}

// ---------------- workspace layout (floats) ----------------
constexpr size_t X_OFF    = 0;
constexpr size_t QKV_OFF  = X_OFF + (size_t)B_ * T_ * DIM;          // 4,194,304
constexpr size_t BUFA_OFF = QKV_OFF + (size_t)B_ * T_ * QKVN;       // +9,437,184
constexpr size_t BUFB_OFF = BUFA_OFF + (size_t)B_ * W_ * CD;        // +1,048,576
constexpr size_t CKT_OFF  = BUFB_OFF + (size_t)B_ * W_ * CD;        // +1,048,576
constexpr size_t CVT_OFF  = CKT_OFF + (size_t)B_ * W_ * DH;
constexpr size_t CK_OFF   = CVT_OFF + (size_t)B_ * W_ * DH;
constexpr size_t CV_OFF   = CK_OFF + (size_t)B_ * NKC * DH;
constexpr size_t GATE_OFF = CV_OFF + (size_t)B_ * NKC * DH;
constexpr size_t SEL_OFF  = GATE_OFF + (size_t)B_ * T_ * 3 * HEADS; // ints here
constexpr size_t MERG_OFF = SEL_OFF + (size_t)B_ * NQ * NSEL;
// total ≈ 24.65M floats ≈ 99 MB

extern "C" void kernel_launch(void* const* d_in, const int* in_sizes, int n_in,
                              void* d_out, int out_size, void* d_ws, size_t ws_size,
                              hipStream_t stream) {
  const float* inp    = (const float*)d_in[0];
  const float* pos    = (const float*)d_in[1];
  const float* norm_w = (const float*)d_in[2];
  const float* Wqkv   = (const float*)d_in[3];
  const float* mem_kv = (const float*)d_in[4];
  const float* kc_W1  = (const float*)d_in[5];
  const float* kc_b1  = (const float*)d_in[6];
  const float* kc_W2  = (const float*)d_in[7];
  const float* kc_b2  = (const float*)d_in[8];
  const float* vc_W1  = (const float*)d_in[9];
  const float* vc_b1  = (const float*)d_in[10];
  const float* vc_W2  = (const float*)d_in[11];
  const float* vc_b2  = (const float*)d_in[12];
  const float* sigma  = (const float*)d_in[13];
  const float* comb_W = (const float*)d_in[14];
  const float* comb_b = (const float*)d_in[15];
  const float* out_W  = (const float*)d_in[16];

  float* ws     = (float*)d_ws;
  float* x      = ws + X_OFF;
  float* qkv    = ws + QKV_OFF;
  float* bufA   = ws + BUFA_OFF;
  float* bufB   = ws + BUFB_OFF;
  float* ckt    = ws + CKT_OFF;
  float* cvt    = ws + CVT_OFF;
  float* ck     = ws + CK_OFF;
  float* cv     = ws + CV_OFF;
  float* gates  = ws + GATE_OFF;
  int*   sel    = (int*)(ws + SEL_OFF);
  float* merged = ws + MERG_OFF;
  float* out    = (float*)d_out;

  const int MROWS = B_ * T_;  // 8192

  // 1. RMSNorm
  rmsnorm_kernel<<<MROWS, 256, 0, stream>>>(inp, norm_w, x);
  // 2. QKV projection: [8192,512] @ [512,1152]
  gemm_wmma<0><<<dim3(QKVN / 64, MROWS / 16), 32, 0, stream>>>(
      x, Wqkv, nullptr, qkv, MROWS, QKVN, DIM);
  // 3. gates
  gates_kernel<<<MROWS, 64, 0, stream>>>(x, comb_W, comb_b, gates);
  // 4-6. key compression MLP
  build_win<<<B_ * W_, 256, 0, stream>>>(qkv, bufA, INNER);
  gemm_wmma<2><<<dim3(CD / 64, (B_ * W_) / 16), 32, 0, stream>>>(
      bufA, kc_W1, kc_b1, bufB, B_ * W_, CD, CD);
  gemm_wmma<1><<<dim3(DH / 64, (B_ * W_) / 16), 32, 0, stream>>>(
      bufB, kc_W2, kc_b2, ckt, B_ * W_, DH, CD);
  // 7-9. value compression MLP
  build_win<<<B_ * W_, 256, 0, stream>>>(qkv, bufA, INNER + DH);
  gemm_wmma<2><<<dim3(CD / 64, (B_ * W_) / 16), 32, 0, stream>>>(
      bufA, vc_W1, vc_b1, bufB, B_ * W_, CD, CD);
  gemm_wmma<1><<<dim3(DH / 64, (B_ * W_) / 16), 32, 0, stream>>>(
      bufB, vc_W2, vc_b2, cvt, B_ * W_, DH, CD);
  // 10. prepend memory kv slot
  pack_ck<<<B_ * NKC, DH, 0, stream>>>(ckt, cvt, mem_kv, ck, cv);
  // 11. compressed attention (writes merged with gate0)
  attn_compressed<<<dim3(T_ / 16, HEADS, B_), 32, 0, stream>>>(
      qkv, ck, cv, gates, merged);
  // 12. importance top-k selection (only rows t<NQ are ever consumed)
  select_topk<<<B_ * NQ, 64, 0, stream>>>(qkv, ck, sel);
  // 13. fine branch (+= gate1)
  attn_fine<<<dim3(NQ * 4, HEADS, B_), 32, 0, stream>>>(qkv, sel, gates, merged);
  // 14. sliding ball branch (+= gate2)
  attn_slide<<<dim3(NB * 16, HEADS, B_), 32, 0, stream>>>(
      qkv, pos, sigma, gates, merged);
  // 15. output projection: [8192,1024] @ [1024,512]
  gemm_wmma<0><<<dim3(DIM / 64, MROWS / 16), 32, 0, stream>>>(
      merged, out_W, nullptr, out, MROWS, DIM, INNER);
}